// PartialDataLoss_38525856645461
// MI455X (gfx1250) — compile-verified
//
#include <hip/hip_runtime.h>

typedef __attribute__((ext_vector_type(2))) float v2f;
typedef __attribute__((ext_vector_type(8))) float v8f;

#define M_PTS 6890
#define N_PTS 50000
#define THRESH 0.1f
#define NTILES (N_PTS / 16)          // 3125, exact
#define WAVES 8
#define BLOCK (WAVES * 32)
#define MAIN_TILES 3120              // 16 * 195: tiles covered by the uniform unroll-2 loop
#define MAIN_ITERS 195

__global__ void pdl_zero_out(float* out) {
    if (threadIdx.x == 0) out[0] = 0.0f;
}

__global__ __launch_bounds__(BLOCK)
void pdl_chamfer_kernel(const float* __restrict__ scan,   // [N_PTS, 3]
                        const float* __restrict__ tmpl,   // [M_PTS, 3]
                        float* __restrict__ out)
{
    __shared__ float tsq_lds[16];
    __shared__ float wave_min[WAVES][16];

    const int tid   = threadIdx.x;
    const int lane  = tid & 31;
    const int wave  = tid >> 5;
    const int mtile = blockIdx.x;

    // ---- Template tile: row m = lane&15 of this 16-row tile ----
    const int mloc = lane & 15;
    const int mrow = mtile * 16 + mloc;
    float tx = 0.0f, ty = 0.0f, tz = 0.0f, tq = 1e30f;  // pad rows -> huge |t|^2 (never passes threshold)
    if (mrow < M_PTS) {
        tx = tmpl[3 * mrow + 0];
        ty = tmpl[3 * mrow + 1];
        tz = tmpl[3 * mrow + 2];
        tq = tx * tx + ty * ty + tz * tz;
    }
    if (wave == 0 && lane < 16) tsq_lds[mloc] = tq;

    // A-matrix 16x4 (f32, 2 VGPRs), pre-scaled by -2 (exact in fp32):
    // lanes 0-15 -> (K0,K1)=(-2x,-2y); lanes 16-31 -> (K2,K3)=(-2z,0)
    const bool lo_half = (lane < 16);
    v2f a;
    a.x = -2.0f * (lo_half ? tx : tz);
    a.y = lo_half ? (-2.0f * ty) : 0.0f;

    __syncthreads();

    // |t|^2 for the 8 rows this lane's C slots cover.
    const int hi = lo_half ? 0 : 8;
    float tqv[8];
#pragma unroll
    for (int v = 0; v < 8; ++v) tqv[v] = tsq_lds[v + hi];

    float mn[8];
#pragma unroll
    for (int v = 0; v < 8; ++v) mn[v] = 1e30f;

    // One 16x16 tile: d2 = WMMA(-2A, B, C = t^2 + s^2); then running min.
    auto process_tile = [&](int jt) {
        const int n = jt * 16 + mloc;
        const float sx = scan[3 * n + 0];
        const float sy = scan[3 * n + 1];
        const float sz = scan[3 * n + 2];
        const float ssq = sx * sx + sy * sy + sz * sz;

        // B-matrix 4x16 (f32, 2 VGPRs): lanes 0-15 -> (x,y); lanes 16-31 -> (z,0)
        v2f b;
        b.x = lo_half ? sx : sz;
        b.y = lo_half ? sy : 0.0f;

        v8f c;
#pragma unroll
        for (int v = 0; v < 8; ++v) c[v] = tqv[v] + ssq;

        c = __builtin_amdgcn_wmma_f32_16x16x4_f32(
                /*neg_a=*/false, a, /*neg_b=*/false, b,
                /*c_mod=*/(short)0, c, /*reuse_a=*/false, /*reuse_b=*/false);

#pragma unroll
        for (int v = 0; v < 8; ++v) mn[v] = fminf(mn[v], c[v]);
    };

    // ---- Main loop: compile-time trip count (195), uniform across all waves.
    // Each wave takes 2 consecutive tiles per iteration -> 2 independent WMMAs for ILP,
    // pure scalar loop control, EXEC stays all-ones.
    const int base0 = wave * 2;
#pragma unroll 1
    for (int it = 0; it < MAIN_ITERS; ++it) {
        const int jt = it * 16 + base0;
        process_tile(jt);
        process_tile(jt + 1);
    }

    // ---- Remainder: tiles 3120..3124 -> waves 0..4 (wave-uniform guard).
    if (wave < NTILES - MAIN_TILES) {
        process_tile(MAIN_TILES + wave);
    }

    // ---- Min across the 16 lanes sharing each template row ----
    // xor masks 8,4,2,1 never cross bit 4: the two 16-lane halves reduce independently.
#pragma unroll
    for (int v = 0; v < 8; ++v) {
        float x = mn[v];
        x = fminf(x, __shfl_xor(x, 8, 32));
        x = fminf(x, __shfl_xor(x, 4, 32));
        x = fminf(x, __shfl_xor(x, 2, 32));
        x = fminf(x, __shfl_xor(x, 1, 32));
        mn[v] = x;
    }

    if (lane == 0) {
#pragma unroll
        for (int v = 0; v < 8; ++v) wave_min[wave][v] = mn[v];       // rows 0..7
    } else if (lane == 16) {
#pragma unroll
        for (int v = 0; v < 8; ++v) wave_min[wave][v + 8] = mn[v];   // rows 8..15
    }
    __syncthreads();

    // ---- Combine waves, threshold, sum, one atomic per block ----
    if (wave == 0) {
        float m = 1e30f;
        if (lane < 16) {
            m = wave_min[0][lane];
#pragma unroll
            for (int w = 1; w < WAVES; ++w) m = fminf(m, wave_min[w][lane]);
        }
        float val = (lane < 16 && m < THRESH) ? m : 0.0f;
        val += __shfl_xor(val, 16, 32);
        val += __shfl_xor(val, 8, 32);
        val += __shfl_xor(val, 4, 32);
        val += __shfl_xor(val, 2, 32);
        val += __shfl_xor(val, 1, 32);
        if (lane == 0) atomicAdd(out, val);
    }
}

extern "C" void kernel_launch(void* const* d_in, const int* in_sizes, int n_in,
                              void* d_out, int out_size, void* d_ws, size_t ws_size,
                              hipStream_t stream) {
    const float* scan = (const float*)d_in[0];   // [50000, 3]
    const float* tmpl = (const float*)d_in[1];   // [6890, 3]
    float* out = (float*)d_out;                  // scalar f32

    pdl_zero_out<<<1, 1, 0, stream>>>(out);

    const int mtiles = (M_PTS + 15) / 16;        // 431 blocks
    pdl_chamfer_kernel<<<mtiles, BLOCK, 0, stream>>>(scan, tmpl, out);
}